// CRNNLoss_38689065402729
// MI455X (gfx1250) — compile-verified
//
#include <hip/hip_runtime.h>
#include <stddef.h>
#include <stdint.h>

// Problem constants (from reference): T=160, B=64, C=6625, S=25
#define T_STEPS   160
#define B_SIZE    64
#define C_CLASSES 6625
#define S_MAX     25
#define L_EXT     51          // 2*S + 1 extended (blank-interleaved) length
#define NEGV      (-1.0e30f)  // finite stand-in for -inf, as in reference

// CDNA5 async global->LDS gather path (gfx1250), probe-confirmed present.
#if defined(__gfx1250__) && __has_builtin(__builtin_amdgcn_global_load_async_to_lds_b32) && __has_builtin(__builtin_amdgcn_s_wait_asynccnt)
#define USE_ASYNC_LDS 1
#else
#define USE_ASYNC_LDS 0
#endif

// Single wave32 per batch element: barriers in a single-wave workgroup are
// S_NOPs on CDNA5, so the 160-step scan has zero barrier overhead; intra-wave
// LDS ordering is enforced by compiler dscnt waits.
__global__ __launch_bounds__(32) void ctc_alpha_kernel(
    const float* __restrict__ pred,        // [T, B, C] raw scores (used as log-probs)
    const int*   __restrict__ targets,     // [B, S] labels in [1, C-1]
    const int*   __restrict__ tgt_len,     // [B] in [5, S]
    float*       __restrict__ per_sample)  // [B] output: per-sample weighted nll
{
    const int b   = blockIdx.x;   // one workgroup (one wave32) per batch element
    const int tid = threadIdx.x;  // 0..31

    __shared__ float s_emit[T_STEPS * L_EXT];   // 32,640 B: all gathered emissions
    __shared__ float s_alpha[2][L_EXT + 2];     // double-buffered alpha, 2-pad for l-1,l-2
    __shared__ int   s_ext[L_EXT];              // extended symbol ids
    __shared__ int   s_skip[L_EXT];             // skip-transition allowed (odd l>=3)

    // ---- build extended target + skip mask (each lane covers l and l+32) --
    for (int l = tid; l < L_EXT; l += 32) {
        int ex = (l & 1) ? targets[b * S_MAX + (l >> 1)] : 0;
        s_ext[l] = ex;
        int sk = 0;
        if ((l & 1) && l >= 3) {
            sk = (targets[b * S_MAX + ((l - 1) >> 1)] !=
                  targets[b * S_MAX + ((l - 3) >> 1)]);
        }
        s_skip[l] = sk;
    }
    if (tid < 2) {  // permanent -inf padding cells at the front of both buffers
        s_alpha[0][tid] = NEGV;
        s_alpha[1][tid] = NEGV;
    }
    __syncthreads();  // S_NOP for single-wave workgroup; kept for portability

    // ---- async gather of all T*L emissions into LDS ----------------------
    // emit[t][l] = pred[(t*B + b)*C + ext[l]]; pipelined on ASYNCcnt.
    {
        const float* rowbase = pred + (size_t)b * C_CLASSES;
        for (int t = 0; t < T_STEPS; ++t) {
            const float* trow = rowbase + (size_t)t * (B_SIZE * C_CLASSES);
            for (int l = tid; l < L_EXT; l += 32) {
                const float* src = trow + s_ext[l];
#if USE_ASYNC_LDS
                __builtin_amdgcn_global_load_async_to_lds_b32(
                    (__attribute__((address_space(1))) int*)src,
                    (__attribute__((address_space(3))) int*)&s_emit[t * L_EXT + l],
                    /*offset=*/0, /*cpol=*/0);
#else
                s_emit[t * L_EXT + l] = *src;
#endif
            }
        }
    }
#if USE_ASYNC_LDS
    __builtin_amdgcn_s_wait_asynccnt(0);
#endif
    __syncthreads();

    // ---- alpha0 ----------------------------------------------------------
    for (int l = tid; l < L_EXT; l += 32) {
        float a0;
        if (l == 0)      a0 = s_emit[0];
        else if (l == 1) a0 = s_emit[1];
        else             a0 = NEGV;
        s_alpha[0][2 + l] = a0;
    }

    // ---- sequential CTC scan over T (double-buffered, no barriers) -------
    int cur = 0;
    for (int t = 1; t < T_STEPS; ++t) {
        const int nxt = cur ^ 1;
        // lane handles l = tid and (if valid) l = tid + 32
        {
            const int l = tid;  // always < 51 for tid < 32
            float a0 = s_alpha[cur][2 + l];
            float a1 = s_alpha[cur][1 + l];
            float a2 = s_skip[l] ? s_alpha[cur][l] : NEGV;
            float m  = fmaxf(a0, fmaxf(a1, a2));
            float v  = m + logf(expf(a0 - m) + expf(a1 - m) + expf(a2 - m));
            s_alpha[nxt][2 + l] = v + s_emit[t * L_EXT + l];
        }
        if (tid + 32 < L_EXT) {
            const int l = tid + 32;
            float a0 = s_alpha[cur][2 + l];
            float a1 = s_alpha[cur][1 + l];
            float a2 = s_skip[l] ? s_alpha[cur][l] : NEGV;
            float m  = fmaxf(a0, fmaxf(a1, a2));
            float v  = m + logf(expf(a0 - m) + expf(a1 - m) + expf(a2 - m));
            s_alpha[nxt][2 + l] = v + s_emit[t * L_EXT + l];
        }
        cur = nxt;
    }

    // ---- finalize: nll = -logaddexp(alpha[2*len], alpha[2*len-1]) --------
    if (tid == 0) {
        const int len = tgt_len[b];
        const int idx = 2 * len;                 // <= 50
        float la = s_alpha[cur][2 + idx];
        float lp = s_alpha[cur][1 + idx];
        float m  = fmaxf(la, lp);
        float nll = -(m + logf(expf(la - m) + expf(lp - m)));
        // zero_infinity=True, then divide by target length (reduction='mean')
        float per = (nll >= 1e29f) ? 0.0f : (nll / (float)len);
        per_sample[b] = per;
    }
}

__global__ __launch_bounds__(32) void ctc_reduce_kernel(
    const float* __restrict__ per_sample, float* __restrict__ out)
{
    if (threadIdx.x == 0) {
        float s = 0.0f;
        for (int i = 0; i < B_SIZE; ++i) s += per_sample[i];
        out[0] = s / (float)B_SIZE;
    }
}

extern "C" void kernel_launch(void* const* d_in, const int* in_sizes, int n_in,
                              void* d_out, int out_size, void* d_ws, size_t ws_size,
                              hipStream_t stream) {
    (void)in_sizes; (void)n_in; (void)out_size; (void)ws_size;
    const float* pred    = (const float*)d_in[0];  // [T,B,C] float32
    const int*   targets = (const int*)  d_in[1];  // [B,S]   int32
    const int*   lens    = (const int*)  d_in[2];  // [B]     int32
    float*       out     = (float*)d_out;          // scalar
    float*       ws      = (float*)d_ws;           // B floats of scratch

    ctc_alpha_kernel<<<B_SIZE, 32, 0, stream>>>(pred, targets, lens, ws);
    ctc_reduce_kernel<<<1, 32, 0, stream>>>(ws, out);
}